// MLP_edge_34514357191071
// MI455X (gfx1250) — compile-verified
//
#include <hip/hip_runtime.h>
#include <hip/hip_bf16.h>

typedef float v2f __attribute__((ext_vector_type(2)));
typedef float v8f __attribute__((ext_vector_type(8)));

// One wave32 handles a tile of 16 edges.
// Layer 1 (dif[16x64] @ W1[64x64]) via 64x V_WMMA_F32_16X16X4_F32 (exact fp32).
// W1 is staged in LDS pre-interleaved as (row k, row k+1) float2 pairs so each
// WMMA B operand is a single conflict-free ds_load_b64 into an aligned pair.
// Layer 2 (bias+ReLU, dot W2) via VALU + cross-lane xor reduction.

#define PAIR_STRIDE 80   // pairs per pair-row; 80*2 DWORDs -> hi half hits banks 32..63

__global__ __launch_bounds__(256) void edge_mlp_wmma_kernel(
    const float* __restrict__ K_h, const float* __restrict__ Q_h,
    const float* __restrict__ P_e, const int* __restrict__ src,
    const int* __restrict__ dst, const float* __restrict__ W1,
    const float* __restrict__ b1, const float* __restrict__ W2,
    const float* __restrict__ b2, float* __restrict__ out, int nEdges)
{
    __shared__ v2f sW1p[32 * PAIR_STRIDE];   // 20 KB of the WGP's 320 KB LDS

    // ---- Stage W1 into LDS as interleaved row pairs (once per block) -------
    // sW1p[kp*STRIDE + n] = { W1[2kp][n], W1[2kp+1][n] }
#pragma unroll
    for (int t = 0; t < 8; ++t) {
        const int i  = threadIdx.x + t * 256;   // 0..2047
        const int kp = i >> 6;                  // pair-row 0..31
        const int n  = i & 63;                  // column
        v2f pr;
        pr.x = W1[(size_t)(2 * kp)     * 64 + n];
        pr.y = W1[(size_t)(2 * kp + 1) * 64 + n];
        sW1p[kp * PAIR_STRIDE + n] = pr;
    }
    __syncthreads();

    const int lane = threadIdx.x & 31;
    const int waveInBlock = threadIdx.x >> 5;
    const int tile = blockIdx.x * (blockDim.x >> 5) + waveInBlock;

    const int n0 = lane & 15;   // row (for A) / column-in-tile (for B,C)
    const int hi = lane >> 4;   // half-wave selector (K pair / row+8)

    // ---- Gather + build A operands in WMMA 16x4 f32 layout -----------------
    int e = tile * 16 + n0;
    if (e >= nEdges) e = nEdges - 1;          // clamp: EXEC must stay all-ones
    const int s = src[e];
    const int d = dst[e];
    const float* __restrict__ kp_ = K_h + (size_t)s * 64;
    const float* __restrict__ qp_ = Q_h + (size_t)d * 64;
    const float* __restrict__ pp_ = P_e + (size_t)s * 64;

    v2f A[16];
#pragma unroll
    for (int kc = 0; kc < 16; ++kc) {
        const int col = 4 * kc + 2 * hi;      // this lane's K pair
        v2f kv = *(const v2f*)(kp_ + col);
        v2f qv = *(const v2f*)(qp_ + col);
        v2f pv = *(const v2f*)(pp_ + col);
        A[kc].x = kv.x - qv.x + pv.x;
        A[kc].y = kv.y - qv.y + pv.y;
    }

    // ---- Layer 1: h[16x64] = dif @ W1, as 4 N-tiles x 16 K-chunks ----------
    v8f acc[4];
#pragma unroll
    for (int nt = 0; nt < 4; ++nt) acc[nt] = (v8f){};

#pragma unroll
    for (int kc = 0; kc < 16; ++kc) {
        // B rows for this lane are k0=4kc+2hi and k0+1 -> pair-row 2kc+hi.
        const v2f* __restrict__ wp = &sW1p[(2 * kc + hi) * PAIR_STRIDE + n0];
#pragma unroll
        for (int nt = 0; nt < 4; ++nt) {
            const v2f b = wp[nt * 16];        // one ds_load_b64, aligned pair
            acc[nt] = __builtin_amdgcn_wmma_f32_16x16x4_f32(
                /*neg_a=*/false, A[kc], /*neg_b=*/false, b,
                /*c_mod=*/(short)0, acc[nt],
                /*reuse_a=*/false, /*reuse_b=*/false);
        }
    }

    // ---- Layer 2: score[M] = sum_n relu(h[M][n]+b1[n]) * W2[n] + b2 --------
    // C layout: VGPR r, lanes 0-15 -> (M=r, N=lane); lanes 16-31 -> (M=r+8).
    float p[8];
#pragma unroll
    for (int r = 0; r < 8; ++r) p[r] = 0.0f;

#pragma unroll
    for (int nt = 0; nt < 4; ++nt) {
        const int n = nt * 16 + n0;           // this lane's column
        const float bb = b1[n];
        const float ww = W2[n];
#pragma unroll
        for (int r = 0; r < 8; ++r) {
            float h = acc[nt][r] + bb;
            h = (h > 0.0f) ? h : 0.0f;        // ReLU
            p[r] += h * ww;
        }
    }

    // Reduce over the 16 columns held within each half-wave (masks 1,2,4,8
    // never cross the 16-lane boundary, so halves reduce independently).
#pragma unroll
    for (int m = 1; m <= 8; m <<= 1) {
#pragma unroll
        for (int r = 0; r < 8; ++r) p[r] += __shfl_xor(p[r], m, 32);
    }

    // Lane with (lane&15) = m < 8 writes row M = m + 8*hi of this tile.
    if (n0 < 8) {
        float val = p[0];
#pragma unroll
        for (int r = 1; r < 8; ++r)           // register select, no scratch
            if (n0 == r) val = p[r];
        const int oe = tile * 16 + n0 + 8 * hi;
        if (oe < nEdges) out[oe] = val + b2[0];
    }
}

extern "C" void kernel_launch(void* const* d_in, const int* in_sizes, int n_in,
                              void* d_out, int out_size, void* d_ws, size_t ws_size,
                              hipStream_t stream) {
    const float* K_h = (const float*)d_in[0];
    const float* Q_h = (const float*)d_in[1];
    const float* P_e = (const float*)d_in[2];
    const int*   src = (const int*)d_in[3];
    const int*   dst = (const int*)d_in[4];
    const float* W1  = (const float*)d_in[5];
    const float* b1  = (const float*)d_in[6];
    const float* W2  = (const float*)d_in[7];
    const float* b2  = (const float*)d_in[8];
    float* out = (float*)d_out;

    const int nEdges = in_sizes[3];                 // src count = 800000
    const int tiles  = (nEdges + 15) / 16;          // 16 edges per wave
    const int wavesPerBlock = 8;                    // 256 threads, wave32
    const int blocks = (tiles + wavesPerBlock - 1) / wavesPerBlock;

    edge_mlp_wmma_kernel<<<blocks, 256, 0, stream>>>(
        K_h, Q_h, P_e, src, dst, W1, b1, W2, b2, out, nEdges);
}